// DeformableTransformerEncoderLayer_37452114821808
// MI455X (gfx1250) — compile-verified
//
#include <hip/hip_runtime.h>
#include <hip/hip_bf16.h>

// Deformable Transformer encoder layer for MI455X (gfx1250).
// fp32 GEMMs on V_WMMA_F32_16X16X4_F32, fed by the Tensor Data Mover
// (tensor_load_to_lds + s_wait_tensorcnt) staging 64x16 / 16x64 tiles in LDS.
// B=4, Lq=8400, D=256, nH=8, hd=32, L=3, P=4, DFF=1024.

#define LQ   8400
#define DM   256
#define NH   8
#define HD   32
#define NLVL 3
#define NPT  4
#define DFF_ 1024

typedef float        v2f __attribute__((ext_vector_type(2)));
typedef float        v8f __attribute__((ext_vector_type(8)));
typedef unsigned int v4u __attribute__((ext_vector_type(4)));
typedef int          v4i __attribute__((ext_vector_type(4)));
typedef int          v8i __attribute__((ext_vector_type(8)));

__device__ __forceinline__ float gelu_exact(float x) {
    return 0.5f * x * (1.0f + erff(x * 0.7071067811865476f));
}

// ------------------------------------------------------------------ TDM load
// 2D tile DMA: copies a tile (tile_d0 x tile_d1) of the tensor at gaddr to LDS at byte offset lds_off,
// elements are 4-byte, row stride = stride0 elements. D# packed per CDNA5 ISA
// 8.3/8.4 (group0: count/lds/global/type, group1: data_size + dims + stride).
// This toolchain exposes the 6-arg builtin: (g0, g1, g2, g3, g4, cpol).
__device__ __forceinline__ void tdm_load_2d(const void* gaddr, unsigned lds_off,
                                            unsigned tile_d0, unsigned tile_d1,
                                            unsigned tens_d0, unsigned tens_d1,
                                            unsigned stride0) {
    unsigned long long ga = (unsigned long long)(uintptr_t)gaddr;
    v4u g0;
    g0.x = 0x1u;                                                  // count=1, user D#
    g0.y = lds_off;                                               // LDS byte address
    g0.z = (unsigned)(ga & 0xFFFFFFFFull);                        // global_addr[31:0]
    g0.w = (unsigned)((ga >> 32) & 0x1FFFFFFull) | (2u << 30);    // [56:32] | type=2
    v8i g1;
    g1[0] = 0x00020000;                                           // data_size=2 -> 4B
    g1[1] = (int)((tens_d0 & 0xFFFFu) << 16);                     // tensor_dim0 lo
    g1[2] = (int)((tens_d0 >> 16) | ((tens_d1 & 0xFFFFu) << 16)); // d0 hi | d1 lo
    g1[3] = (int)((tens_d1 >> 16) | (tile_d0 << 16));             // d1 hi | tile_dim0
    g1[4] = (int)tile_d1;                                         // tile_dim1, dim2=0
    g1[5] = (int)stride0;                                         // dim0_stride[31:0]
    g1[6] = 0;                                                    // stride hi, dim1_stride lo
    g1[7] = 0;
    v4i z4 = {0, 0, 0, 0};
    v8i z8 = {0, 0, 0, 0, 0, 0, 0, 0};
    __builtin_amdgcn_tensor_load_to_lds(g0, g1, z4, z4, z8, 0);
}

// ---------------------------------------------------------------- elementwise
__global__ __launch_bounds__(256) void add_kernel(const float* __restrict__ a,
                                                  const float* __restrict__ b,
                                                  float* __restrict__ c, int n) {
    int i = blockIdx.x * blockDim.x + threadIdx.x;
    if (i < n) c[i] = a[i] + b[i];
}

// --------------------------------------------- TDM-fed LDS-tiled WMMA GEMM
// C[M,N] = act(A[M,K] @ W[K,N] + bias[N]). 64x64 macro-tile per 256-thread
// workgroup (8 waves, each a 16x32 strip = 2 WMMA tiles). Requires M%64==0,
// N%64==0, K%16==0; grid = (N/64, M/64) fully covered -> EXEC all-ones.
template <int ACT>  // 0 = none, 1 = exact GELU
__global__ __launch_bounds__(256) void wmma_gemm_tdm(const float* __restrict__ A,
                                                     const float* __restrict__ W,
                                                     const float* __restrict__ bias,
                                                     float* __restrict__ C,
                                                     int M, int N, int K) {
    __shared__ float smem[64 * 16 + 16 * 64];   // shA then shB (HIP LDS base = 0)
    float* shA = smem;                           // [64][16], LDS byte off 0
    float* shB = smem + 64 * 16;                 // [16][64], LDS byte off 4096

    const int lane = threadIdx.x & 31;
    const int wave = threadIdx.x >> 5;
    const int lm   = lane & 15;
    const int koff = (lane >> 4) << 1;           // 0 | 2
    const int wr   = (wave >> 1) << 4;           // wave row in macro-tile: 0..48
    const int wc   = (wave & 1) << 5;            // wave col in macro-tile: 0 | 32
    const int rowB = blockIdx.y << 6;
    const int colB = blockIdx.x << 6;

    v8f acc0 = {}, acc1 = {};
    for (int k0 = 0; k0 < K; k0 += 16) {
        if (wave == 0) {
            // A slab: 64 rows x 16 K (row-contiguous 16 floats), stride K
            tdm_load_2d(A + (size_t)rowB * K + k0, 0u, 16u, 64u,
                        (unsigned)K, (unsigned)M, (unsigned)K);
            // B slab: 16 K-rows x 64 cols (row-contiguous 64 floats), stride N
            tdm_load_2d(W + (size_t)k0 * N + colB, 4096u, 64u, 16u,
                        (unsigned)N, (unsigned)K, (unsigned)N);
            __builtin_amdgcn_s_wait_tensorcnt((short)0);
        }
        __syncthreads();
#pragma unroll
        for (int kk = 0; kk < 16; kk += 4) {
            v2f a = *(const v2f*)&shA[(wr + lm) * 16 + kk + koff];
            v2f b0, b1;
            b0.x = shB[(kk + koff) * 64 + wc + lm];
            b0.y = shB[(kk + koff + 1) * 64 + wc + lm];
            b1.x = shB[(kk + koff) * 64 + wc + 16 + lm];
            b1.y = shB[(kk + koff + 1) * 64 + wc + 16 + lm];
            acc0 = __builtin_amdgcn_wmma_f32_16x16x4_f32(false, a, false, b0,
                                                         (short)0, acc0, false, false);
            acc1 = __builtin_amdgcn_wmma_f32_16x16x4_f32(false, a, false, b1,
                                                         (short)0, acc1, false, false);
        }
        __syncthreads();
    }

    const int rhi  = (lane >> 4) << 3;           // +8 rows for high half-wave
    const int row0 = rowB + wr + rhi;
    const int c0   = colB + wc + lm;
    const int c1   = c0 + 16;
    const float bv0 = bias[c0], bv1 = bias[c1];
#pragma unroll
    for (int r = 0; r < 8; ++r) {
        float v0 = acc0[r] + bv0;
        float v1 = acc1[r] + bv1;
        if (ACT == 1) { v0 = gelu_exact(v0); v1 = gelu_exact(v1); }
        C[(size_t)(row0 + r) * N + c0] = v0;
        C[(size_t)(row0 + r) * N + c1] = v1;
    }
}

// ------------------------------------------- direct-load WMMA GEMM (N=96 case)
__global__ __launch_bounds__(256) void wmma_gemm_direct(const float* __restrict__ A,
                                                        const float* __restrict__ W,
                                                        const float* __restrict__ bias,
                                                        float* __restrict__ C,
                                                        int M, int N, int K) {
    const int lane   = threadIdx.x & 31;
    const int wave   = threadIdx.x >> 5;
    const int tilesN = N >> 4;
    const int tiles  = (M >> 4) * tilesN;
    const int t      = blockIdx.x * 8 + wave;
    if (t >= tiles) return;  // wave-uniform

    const int row0 = (t / tilesN) << 4;
    const int col0 = (t % tilesN) << 4;
    const int lm   = lane & 15;
    const int koff = (lane >> 4) << 1;

    v8f acc = {};
    const float* Arow = A + (size_t)(row0 + lm) * K;
    const float* Wcol = W + col0 + lm;
    for (int k = 0; k < K; k += 4) {
        v2f a = *(const v2f*)(Arow + k + koff);
        const float* wp = Wcol + (size_t)(k + koff) * N;
        v2f b;
        b.x = wp[0];
        b.y = wp[N];
        acc = __builtin_amdgcn_wmma_f32_16x16x4_f32(false, a, false, b,
                                                    (short)0, acc, false, false);
    }
    const int   col = col0 + lm;
    const float bv  = bias[col];
    const int   rhi = (lane >> 4) << 3;
#pragma unroll
    for (int r = 0; r < 8; ++r)
        C[(size_t)(row0 + rhi + r) * N + col] = acc[r] + bv;
}

// ------------------------------------------------------------------- softmax12
__global__ __launch_bounds__(256) void softmax12(float* __restrict__ aw, int rows) {
    int r = blockIdx.x * blockDim.x + threadIdx.x;
    if (r >= rows) return;
    float* p = aw + (size_t)r * (NLVL * NPT);
    float m = -3.4e38f;
#pragma unroll
    for (int i = 0; i < NLVL * NPT; ++i) m = fmaxf(m, p[i]);
    float e[NLVL * NPT];
    float s = 0.f;
#pragma unroll
    for (int i = 0; i < NLVL * NPT; ++i) { e[i] = __expf(p[i] - m); s += e[i]; }
    const float inv = 1.0f / s;
#pragma unroll
    for (int i = 0; i < NLVL * NPT; ++i) p[i] = e[i] * inv;
}

// --------------------------------------------------------- deformable sampling
// One wave per (b, q, h); lane = channel d. value layout [B,Lq,NH,HD].
__global__ __launch_bounds__(256) void deform_attn(const float* __restrict__ value,
                                                   const float* __restrict__ refp,
                                                   const float* __restrict__ off,
                                                   const float* __restrict__ aw,
                                                   float* __restrict__ out, int B) {
    const int Hs[NLVL] = {80, 40, 20};
    const int Ws[NLVL] = {80, 40, 20};
    const int S0[NLVL] = {0, 6400, 8000};

    const int gid = blockIdx.x * blockDim.x + threadIdx.x;
    const int d   = gid & 31;
    const int w   = gid >> 5;
    const int h   = w & 7;
    const int q   = (w >> 3) % LQ;
    const int b   = w / (NH * LQ);
    if (b >= B) return;

    const size_t bq   = (size_t)b * LQ + q;
    const float* offp = off + (bq * NH + h) * (NLVL * NPT * 2);
    const float* awp  = aw + (bq * NH + h) * (NLVL * NPT);

    float acc = 0.f;
#pragma unroll
    for (int l = 0; l < NLVL; ++l) {
        const int    Hh    = Hs[l], Ww = Ws[l];
        const float  rx    = refp[(bq * NLVL + l) * 2 + 0];
        const float  ry    = refp[(bq * NLVL + l) * 2 + 1];
        const float* vbase = value + (((size_t)b * LQ + S0[l]) * NH + h) * HD + d;
#pragma unroll
        for (int p = 0; p < NPT; ++p) {
            const float lx = rx + offp[(l * NPT + p) * 2 + 0] / (float)Ww;
            const float ly = ry + offp[(l * NPT + p) * 2 + 1] / (float)Hh;
            const float x  = lx * (float)Ww - 0.5f;
            const float y  = ly * (float)Hh - 0.5f;
            const float x0 = floorf(x), y0 = floorf(y);
            const float fx = x - x0, fy = y - y0;
            const int   xi = (int)x0, yi = (int)y0;
            float sample = 0.f;
#pragma unroll
            for (int dy = 0; dy < 2; ++dy) {
#pragma unroll
                for (int dx = 0; dx < 2; ++dx) {
                    const int xc = xi + dx, yc = yi + dy;
                    if (xc >= 0 && xc < Ww && yc >= 0 && yc < Hh) {
                        const float wgt = (dx ? fx : 1.f - fx) * (dy ? fy : 1.f - fy);
                        sample += wgt * vbase[(size_t)(yc * Ww + xc) * (NH * HD)];
                    }
                }
            }
            acc += awp[l * NPT + p] * sample;
        }
    }
    out[bq * DM + h * HD + d] = acc;
}

// ------------------------------------------------- residual add + layer norm
__global__ __launch_bounds__(256) void add_layernorm(const float* __restrict__ x,
                                                     const float* __restrict__ y,
                                                     const float* __restrict__ g,
                                                     const float* __restrict__ bt,
                                                     float* __restrict__ out, int rows) {
    const int lane = threadIdx.x & 31;
    const int row  = blockIdx.x * 8 + (threadIdx.x >> 5);
    if (row >= rows) return;
    const float* xp = x + (size_t)row * DM;
    const float* yp = y + (size_t)row * DM;
    float v[8];
    float s = 0.f;
#pragma unroll
    for (int i = 0; i < 8; ++i) {
        const int c = lane + i * 32;
        v[i] = xp[c] + yp[c];
        s += v[i];
    }
#pragma unroll
    for (int m = 16; m > 0; m >>= 1) s += __shfl_xor(s, m, 32);
    const float mean = s * (1.0f / DM);
    float ss = 0.f;
#pragma unroll
    for (int i = 0; i < 8; ++i) {
        const float dd = v[i] - mean;
        ss += dd * dd;
    }
#pragma unroll
    for (int m = 16; m > 0; m >>= 1) ss += __shfl_xor(ss, m, 32);
    const float inv = rsqrtf(ss * (1.0f / DM) + 1e-5f);
    float* op = out + (size_t)row * DM;
#pragma unroll
    for (int i = 0; i < 8; ++i) {
        const int c = lane + i * 32;
        op[c] = (v[i] - mean) * inv * g[c] + bt[c];
    }
}

// ----------------------------------------------------------------------- host
static inline int cdiv(int a, int b) { return (a + b - 1) / b; }

extern "C" void kernel_launch(void* const* d_in, const int* in_sizes, int n_in,
                              void* d_out, int out_size, void* d_ws, size_t ws_size,
                              hipStream_t stream) {
    const float* src  = (const float*)d_in[0];
    const float* refp = (const float*)d_in[1];
    const float* pos  = (const float*)d_in[2];
    // d_in[3] spatial_shapes / d_in[4] level_start_index: int64 constants, hardcoded
    const float* off_w = (const float*)d_in[5];
    const float* off_b = (const float*)d_in[6];
    const float* aw_w  = (const float*)d_in[7];
    const float* aw_b  = (const float*)d_in[8];
    const float* val_w = (const float*)d_in[9];
    const float* val_b = (const float*)d_in[10];
    const float* out_w = (const float*)d_in[11];
    const float* out_b = (const float*)d_in[12];
    const float* l1w   = (const float*)d_in[13];
    const float* l1b   = (const float*)d_in[14];
    const float* l2w   = (const float*)d_in[15];
    const float* l2b   = (const float*)d_in[16];
    const float* ln1g  = (const float*)d_in[17];
    const float* ln1b  = (const float*)d_in[18];
    const float* ln2g  = (const float*)d_in[19];
    const float* ln2b  = (const float*)d_in[20];

    const int B = in_sizes[0] / (LQ * DM);  // 4
    const int M = B * LQ;                   // 33600 (divisible by 64)

    float* ws = (float*)d_ws;
    size_t o  = 0;
    float* q    = ws + o; o += (size_t)M * DM;
    float* val  = ws + o; o += (size_t)M * DM;
    float* offv = ws + o; o += (size_t)M * (NH * NLVL * NPT * 2);
    float* awv  = ws + o; o += (size_t)M * (NH * NLVL * NPT);
    float* attn = ws + o; o += (size_t)M * DM;
    float* h    = ws + o; o += (size_t)M * DM;
    float* f1   = ws + o; o += (size_t)M * DFF_;
    float* proj = q;    // q dead after aw GEMM
    float* f2   = val;  // value dead after sampling

    // 1) q = src + pos
    add_kernel<<<cdiv(M * DM, 256), 256, 0, stream>>>(src, pos, q, M * DM);

    // 2) value = src @ val_w + val_b          [M,256]
    wmma_gemm_tdm<0><<<dim3(DM / 64, M / 64), 256, 0, stream>>>(src, val_w, val_b, val, M, DM, DM);
    // 3) offsets = q @ off_w + off_b          [M,192]
    wmma_gemm_tdm<0><<<dim3(192 / 64, M / 64), 256, 0, stream>>>(q, off_w, off_b, offv, M, 192, DM);
    // 4) aw = q @ aw_w + aw_b                 [M,96]  (96 % 64 != 0 -> direct path)
    wmma_gemm_direct<<<cdiv((M / 16) * (96 / 16), 8), 256, 0, stream>>>(q, aw_w, aw_b, awv, M, 96, DM);

    // 5) softmax over 12 (level,point) logits per (b,q,h)
    softmax12<<<cdiv(M * NH, 256), 256, 0, stream>>>(awv, M * NH);

    // 6) deformable bilinear sampling -> attn [M,256]
    deform_attn<<<cdiv(M * NH * HD, 256), 256, 0, stream>>>(val, refp, offv, awv, attn, B);

    // 7) proj = attn @ out_w + out_b          [M,256]
    wmma_gemm_tdm<0><<<dim3(DM / 64, M / 64), 256, 0, stream>>>(attn, out_w, out_b, proj, M, DM, DM);

    // 8) h = LN(src + proj)
    add_layernorm<<<cdiv(M, 8), 256, 0, stream>>>(src, proj, ln1g, ln1b, h, M);

    // 9) f1 = gelu(h @ lin1_w + lin1_b)       [M,1024]
    wmma_gemm_tdm<1><<<dim3(DFF_ / 64, M / 64), 256, 0, stream>>>(h, l1w, l1b, f1, M, DFF_, DM);
    // 10) f2 = f1 @ lin2_w + lin2_b           [M,256]
    wmma_gemm_tdm<0><<<dim3(DM / 64, M / 64), 256, 0, stream>>>(f1, l2w, l2b, f2, M, DM, DFF_);

    // 11) out = LN(h + f2)
    add_layernorm<<<cdiv(M, 8), 256, 0, stream>>>(h, f2, ln2g, ln2b, (float*)d_out, M);
}